// _ScaledDotProductAttention_29678224016108
// MI455X (gfx1250) — compile-verified
//
#include <hip/hip_runtime.h>

// Scaled dot-product attention for MI455X (gfx1250), fp32 via V_WMMA_F32_16X16X4_F32.
// B=8, H=16, T=1024, D=64. Outputs: out [B,H,T,D] then attn_weights [B,H,T,T], concatenated.
//
// Roofline: ~34 GFLOP vs ~670 MB mandatory HBM traffic -> bandwidth-bound on the
// 537 MB attn-weights write. Weights/output stores are non-temporal so the Q/K/V
// working set (~100 MB) stays resident in the 192 MB L2 across row-tile re-reads.

#define TT    1024          // sequence length
#define DD    64            // head dim
#define NBH   128           // B*H
#define SCALE 0.125f        // 1/sqrt(64)
#define SROW  (TT + 4)      // padded LDS score row stride (floats): bank-conflict-free
#define QROW  (DD + 4)      // padded LDS Q row stride (floats)

typedef __attribute__((ext_vector_type(2))) float v2f;
typedef __attribute__((ext_vector_type(4))) float v4f;
typedef __attribute__((ext_vector_type(8))) float v8f;

__launch_bounds__(256)
__global__ void sdpa_f32_wmma_kernel(const float* __restrict__ q,
                                     const float* __restrict__ k,   // [BH, D, T] (pre-transposed)
                                     const float* __restrict__ v,   // [BH, T, D]
                                     const float* __restrict__ mask,// [B, T]
                                     float* __restrict__ out,       // [BH, T, D]
                                     float* __restrict__ wts) {     // [BH, T, T]
    __shared__ float lds_q[16 * QROW];   //  4352 B : Q tile 16x64 (padded)
    __shared__ float lds_m[TT];          //  4096 B : mask row
    __shared__ float lds_s[16 * SROW];   // 65792 B : scores / probabilities 16x1024 (padded)
    __shared__ float lds_o[16 * DD];     //  4096 B : output accumulator 16x64

    const int tid  = threadIdx.x;        // 0..255, 8 waves of 32
    const int wave = tid >> 5;
    const int lane = tid & 31;
    const int rt   = blockIdx.x;         // row tile 0..63
    const int bh   = blockIdx.y;         // 0..127
    const int b    = bh >> 4;            // H = 16
    const int row0 = rt * 16;

    const float* qb = q + (size_t)bh * TT * DD + (size_t)row0 * DD;
    const float* kb = k + (size_t)bh * DD * TT;
    const float* vb = v + (size_t)bh * TT * DD;
    float*       ob = out + (size_t)bh * TT * DD + (size_t)row0 * DD;
    float*       wb = wts + (size_t)bh * TT * TT + (size_t)row0 * TT;

    // ---- stage Q tile, mask row; zero output accumulator ----
    {
        const int r  = tid >> 4;
        const int c4 = (tid & 15) << 2;
        const v4f qv = *(const v4f*)(qb + r * DD + c4);
        *(v4f*)(&lds_q[r * QROW + c4]) = qv;

        *(v4f*)(&lds_m[tid << 2]) = *(const v4f*)(mask + (size_t)b * TT + (tid << 2));

        v4f z = {0.f, 0.f, 0.f, 0.f};
        *(v4f*)(&lds_o[tid << 2]) = z;
    }
    __syncthreads();

    const int m  = lane & 15;            // C-tile column lane / A-matrix row
    const int lh = lane >> 4;            // selects K pair {2lh, 2lh+1} of the x4 step

    // ---- preload A operand (Q) into registers: reused across all 8 column tiles ----
    v2f qa[16];
#pragma unroll
    for (int ks = 0; ks < 16; ++ks) {
        const int d0 = ks * 4 + lh * 2;
        qa[ks] = *(const v2f*)(&lds_q[m * QROW + d0]);
    }

    // ---- phase 1: scores tile row = Q (16x64) x K (64x1024); wave w covers cols [w*128, w*128+128) ----
#pragma unroll 1
    for (int t = 0; t < 8; ++t) {
        const int col0 = (wave * 8 + t) * 16 + m;   // this lane's column
        v8f c = {};
#pragma unroll
        for (int ks = 0; ks < 16; ++ks) {
            const int d0 = ks * 4 + lh * 2;
            v2f bm;
            bm.x = kb[(size_t)d0 * TT + col0];
            bm.y = kb[(size_t)(d0 + 1) * TT + col0];
            c = __builtin_amdgcn_wmma_f32_16x16x4_f32(false, qa[ks], false, bm,
                                                      (short)0, c, false, false);
        }
#pragma unroll
        for (int i = 0; i < 8; ++i)
            lds_s[(i + lh * 8) * SROW + col0] = c[i];
    }
    __syncthreads();

    // ---- phase 2: fused scale + outer-product mask + softmax (16 threads per row) ----
    {
        const int   srow = tid >> 4;     // 0..15
        const int   sc   = tid & 15;
        const float mi   = lds_m[row0 + srow];
        float* prow = &lds_s[srow * SROW];
        float* wrow = wb + (size_t)srow * TT;

        float mx = -3.402823466e38f;
#pragma unroll 4
        for (int t = 0; t < 16; ++t) {
            const int f = (sc + t * 16) << 2;           // float offset (16B aligned)
            v4f s4 = *(v4f*)(prow + f);
            const v4f mj = *(const v4f*)(&lds_m[f]);
            s4.x = s4.x * SCALE + mi * mj.x;
            s4.y = s4.y * SCALE + mi * mj.y;
            s4.z = s4.z * SCALE + mi * mj.z;
            s4.w = s4.w * SCALE + mi * mj.w;
            *(v4f*)(prow + f) = s4;
            mx = fmaxf(mx, fmaxf(fmaxf(s4.x, s4.y), fmaxf(s4.z, s4.w)));
        }
        mx = fmaxf(mx, __shfl_xor(mx, 1));
        mx = fmaxf(mx, __shfl_xor(mx, 2));
        mx = fmaxf(mx, __shfl_xor(mx, 4));
        mx = fmaxf(mx, __shfl_xor(mx, 8));   // stays within 16-lane half (wave32)

        float sum = 0.f;
#pragma unroll 4
        for (int t = 0; t < 16; ++t) {
            const int f = (sc + t * 16) << 2;
            v4f s4 = *(v4f*)(prow + f);
            s4.x = __expf(s4.x - mx);
            s4.y = __expf(s4.y - mx);
            s4.z = __expf(s4.z - mx);
            s4.w = __expf(s4.w - mx);
            *(v4f*)(prow + f) = s4;
            sum += (s4.x + s4.y) + (s4.z + s4.w);
        }
        sum += __shfl_xor(sum, 1);
        sum += __shfl_xor(sum, 2);
        sum += __shfl_xor(sum, 4);
        sum += __shfl_xor(sum, 8);
        const float inv = 1.0f / sum;

#pragma unroll 4
        for (int t = 0; t < 16; ++t) {
            const int f = (sc + t * 16) << 2;
            v4f s4 = *(v4f*)(prow + f);
            s4 *= inv;
            *(v4f*)(prow + f) = s4;                    // keep P for the PV matmul
            // weights are write-once/never-read: non-temporal store (TH=NT) so the
            // 537 MB stream does not evict the L2-resident K/V working set.
            __builtin_nontemporal_store(s4, (v4f*)(wrow + f));
        }
    }
    __syncthreads();

    // ---- phase 3: out tile (16x64) = P (16x1024) x V (1024x64) ----
    // wave -> (output col tile ot = wave&3, k-half kh = wave>>2); partials merged via ds_add_f32.
    {
        const int ot    = wave & 3;
        const int kh    = wave >> 2;
        const int vcol  = ot * 16 + m;
        const int kbase = kh * 512;
        v8f c = {};
#pragma unroll 4
        for (int s = 0; s < 128; ++s) {
            const int k0 = kbase + s * 4 + lh * 2;
            const v2f a  = *(const v2f*)(&lds_s[m * SROW + k0]);
            v2f bm;
            bm.x = vb[(size_t)k0 * DD + vcol];
            bm.y = vb[(size_t)(k0 + 1) * DD + vcol];
            c = __builtin_amdgcn_wmma_f32_16x16x4_f32(false, a, false, bm,
                                                      (short)0, c, false, false);
        }
#pragma unroll
        for (int i = 0; i < 8; ++i)
            atomicAdd(&lds_o[(i + lh * 8) * DD + vcol], c[i]);
    }
    __syncthreads();

    // ---- write output tile: 16x64 contiguous floats (streaming, non-temporal) ----
    {
        const v4f o4 = *(const v4f*)(&lds_o[tid << 2]);
        __builtin_nontemporal_store(o4, (v4f*)(ob + (tid << 2)));
    }
}

extern "C" void kernel_launch(void* const* d_in, const int* in_sizes, int n_in,
                              void* d_out, int out_size, void* d_ws, size_t ws_size,
                              hipStream_t stream) {
    const float* q    = (const float*)d_in[0];
    const float* k    = (const float*)d_in[1];
    const float* v    = (const float*)d_in[2];
    const float* mask = (const float*)d_in[3];

    float* out = (float*)d_out;                       // [B,H,T,D]
    float* wts = out + (size_t)NBH * TT * DD;         // [B,H,T,T] follows

    dim3 grid(TT / 16, NBH);
    sdpa_f32_wmma_kernel<<<grid, 256, 0, stream>>>(q, k, v, mask, out, wts);
}